// SparseMatrixEquivariantLayerBlock_76132590289315
// MI455X (gfx1250) — compile-verified
//
#include <hip/hip_runtime.h>

typedef float v2f __attribute__((ext_vector_type(2)));
typedef float v8f __attribute__((ext_vector_type(8)));

#define NDIM  64
#define NROWS 50000   // N in the reference

__device__ __forceinline__ v8f wmma_f32_16x16x4(v2f a, v2f b, v8f c) {
    // (neg_a, A, neg_b, B, c_mod, C, reuse_a, reuse_b)
    return __builtin_amdgcn_wmma_f32_16x16x4_f32(false, a, false, b, (short)0, c,
                                                 false, false);
}

// ---------------------------------------------------------------- zero scratch
__global__ void zero_f32(float* __restrict__ p, long long n) {
    long long i = (long long)blockIdx.x * blockDim.x + threadIdx.x;
    if (i < n) p[i] = 0.0f;
}

// ------------------------------------------------- pooling: P0/P1/G scatter-add
__global__ void pool_kernel(const float* __restrict__ x,
                            const int* __restrict__ xr, const int* __restrict__ xc,
                            float* __restrict__ P0, float* __restrict__ P1,
                            float* __restrict__ G, long long total) {
    long long gid = (long long)blockIdx.x * blockDim.x + threadIdx.x;
    if (gid >= total) return;
    long long e = gid >> 6;
    int f = (int)(gid & 63);
    int r = xr[e], c = xc[e];
    float v = x[gid];
    atomicAdd(&P0[(size_t)r * NDIM + f], v);   // pool over axis 0 -> keyed by idx[0]
    atomicAdd(&P1[(size_t)c * NDIM + f], v);   // pool over axis 1 -> keyed by idx[1]
    if (r == c) atomicAdd(&G[(size_t)r * NDIM + f], v);
}

// --------------------------------- fused GEMM: U{0,1,D} = P1@Wa + P0@Wb + G@Wc
__global__ void gemm_u_kernel(const float* __restrict__ P1, const float* __restrict__ P0,
                              const float* __restrict__ G,  const float* __restrict__ W,
                              float* __restrict__ U) {
    const int wop[3][3] = { {2, 4, 6},    // U0: (p_1,b_0)(p_0,b_0)(g_d,b_0)
                            {3, 5, 7},    // U1: (p_1,b_1)(p_0,b_1)(g_d,b_1)
                            {8, 9, 10} }; // UD: (p_1,e_d)(p_0,e_d)(g_d,e_d)
    int tileRow = blockIdx.x * 16;
    int u    = blockIdx.y;
    int lane = threadIdx.x & 31;
    int wave = threadIdx.x >> 5;          // 4 waves -> 4 chunks of 16 output cols
    int m    = lane & 15;                 // A-matrix row for this lane
    int kh   = lane >> 4;                 // K-pair select (ISA 16x4 A layout)
    int ncol = wave * 16 + (lane & 15);   // B/D column for this lane

    const float* srcs[3] = { P1, P0, G };
    v8f acc = {0.f, 0.f, 0.f, 0.f, 0.f, 0.f, 0.f, 0.f};
    for (int s = 0; s < 3; ++s) {
        const float* A = srcs[s] + (size_t)tileRow * NDIM;
        const float* B = W + (size_t)wop[u][s] * NDIM * NDIM;
#pragma unroll
        for (int k0 = 0; k0 < NDIM; k0 += 4) {
            int ka = k0 + kh * 2;
            v2f a, b;
            a.x = A[m * NDIM + ka];
            a.y = A[m * NDIM + ka + 1];
            b.x = B[ka * NDIM + ncol];
            b.y = B[(ka + 1) * NDIM + ncol];
            acc = wmma_f32_16x16x4(a, b, acc);
        }
    }
    // D layout: lane>>4 selects row half (M 0-7 vs 8-15), vgpr v = M&7
    float* Uo = U + (size_t)u * NROWS * NDIM;
#pragma unroll
    for (int v = 0; v < 8; ++v) {
        int row = tileRow + kh * 8 + v;
        Uo[(size_t)row * NDIM + ncol] = acc[v];
    }
}

// ------------------- finalize: Y = U0[y0] + U1[y1] + (y0==y1)*UD[y0] + sum(bias)
__global__ void finalize_kernel(const int* __restrict__ yr, const int* __restrict__ yc,
                                const float* __restrict__ U, const float* __restrict__ bias,
                                int nOps, float* __restrict__ Y, long long total) {
    __shared__ float bs;
    if (threadIdx.x == 0) {
        float s = 0.0f;
        for (int i = 0; i < nOps; ++i) s += bias[i];
        bs = s;
    }
    __syncthreads();
    long long gid = (long long)blockIdx.x * blockDim.x + threadIdx.x;
    if (gid >= total) return;
    long long e = gid >> 6;
    int f = (int)(gid & 63);
    int y0 = yr[e], y1 = yc[e];
    const float* U0 = U;
    const float* U1 = U + (size_t)NROWS * NDIM;
    const float* UD = U + 2 * (size_t)NROWS * NDIM;
    float v = U0[(size_t)y0 * NDIM + f] + U1[(size_t)y1 * NDIM + f] + bs;
    if (y0 == y1) v += UD[(size_t)y0 * NDIM + f];
    Y[gid] = v;
}

// ---------------- intersections: Y[out[j]] += (x[in[j]] @ W{0|1}) via WMMA
__global__ void intersect_kernel(const float* __restrict__ x,
                                 const int* __restrict__ id_in, const int* __restrict__ id_out,
                                 const int* __restrict__ tr_in, const int* __restrict__ tr_out,
                                 const float* __restrict__ W, float* __restrict__ Y,
                                 int nId, int nTr) {
    int op = blockIdx.y;                  // 0 = identity, 1 = transpose
    int count = op ? nTr : nId;
    const int* gin  = op ? tr_in  : id_in;
    const int* gout = op ? tr_out : id_out;
    int tileRow = blockIdx.x * 16;
    if (tileRow >= count) return;         // wave-uniform exit (EXEC stays full)

    int lane = threadIdx.x & 31;
    int wave = threadIdx.x >> 5;
    int m    = lane & 15;
    int kh   = lane >> 4;
    int ncol = wave * 16 + (lane & 15);

    int ar = tileRow + m;
    if (ar >= count) ar = count - 1;      // clamp (no divergence before WMMA)
    long long srow = gin[ar];
    const float* A = x + srow * NDIM;
    const float* B = W + (size_t)op * NDIM * NDIM;

    v8f acc = {0.f, 0.f, 0.f, 0.f, 0.f, 0.f, 0.f, 0.f};
#pragma unroll
    for (int k0 = 0; k0 < NDIM; k0 += 4) {
        int ka = k0 + kh * 2;
        v2f a, b;
        a.x = A[ka];
        a.y = A[ka + 1];
        b.x = B[ka * NDIM + ncol];
        b.y = B[(ka + 1) * NDIM + ncol];
        acc = wmma_f32_16x16x4(a, b, acc);
    }
#pragma unroll
    for (int v = 0; v < 8; ++v) {
        int orow = tileRow + kh * 8 + v;
        if (orow < count) {
            long long dst = gout[orow];
            atomicAdd(&Y[dst * NDIM + ncol], acc[v]);
        }
    }
}

// -------------------------------------------------------------------- launcher
extern "C" void kernel_launch(void* const* d_in, const int* in_sizes, int n_in,
                              void* d_out, int out_size, void* d_ws, size_t ws_size,
                              hipStream_t stream) {
    const float* x_vals  = (const float*)d_in[0];
    const int*   x_idx   = (const int*)  d_in[1];
    const int*   y_idx   = (const int*)  d_in[2];
    const int*   id_in   = (const int*)  d_in[3];
    const int*   id_out  = (const int*)  d_in[4];
    const int*   tr_in   = (const int*)  d_in[5];
    const int*   tr_out  = (const int*)  d_in[6];
    const float* weights = (const float*)d_in[7];
    const float* bias    = (const float*)d_in[8];
    float* Y = (float*)d_out;

    const long long nnzX = (long long)in_sizes[1] / 2;   // x_idx is [2, nnz]
    const long long nnzY = (long long)in_sizes[2] / 2;   // y_idx is [2, nnz]
    const int nId  = in_sizes[3];
    const int nTr  = in_sizes[5];
    const int nOps = in_sizes[8];

    float* ws = (float*)d_ws;
    const size_t tab = (size_t)NROWS * NDIM;  // one [N,64] table
    float* P0 = ws;
    float* P1 = ws + tab;
    float* G  = ws + 2 * tab;
    float* U  = ws + 3 * tab;                 // U0, U1, UD contiguous

    // 1) zero the pooled tables (atomics accumulate into them)
    long long nz = 3 * (long long)tab;
    zero_f32<<<(unsigned)((nz + 255) / 256), 256, 0, stream>>>(ws, nz);

    // 2) pooling scatter-add pass over all nnz*64 elements
    long long totalX = nnzX * NDIM;
    pool_kernel<<<(unsigned)((totalX + 255) / 256), 256, 0, stream>>>(
        x_vals, x_idx, x_idx + nnzX, P0, P1, G, totalX);

    // 3) fused WMMA GEMMs -> U0, U1, UD  (tables live in L2: 3 x 12.8 MB)
    gemm_u_kernel<<<dim3(NROWS / 16, 3), 128, 0, stream>>>(P1, P0, G, weights, U);

    // 4) finalize base output (gathers hit L2) + summed bias
    long long totalY = nnzY * NDIM;
    finalize_kernel<<<(unsigned)((totalY + 255) / 256), 256, 0, stream>>>(
        y_idx, y_idx + nnzY, U, bias, nOps, Y, totalY);

    // 5) identity + transpose intersections, WMMA + atomic scatter-add
    int maxCnt = nId > nTr ? nId : nTr;
    intersect_kernel<<<dim3((maxCnt + 15) / 16, 2), 128, 0, stream>>>(
        x_vals, id_in, id_out, tr_in, tr_out, weights, Y, nId, nTr);
}